// SQuaReFuse_4028679323666
// MI455X (gfx1250) — compile-verified
//
#include <hip/hip_runtime.h>
#include <hip/hip_bf16.h>
#include <math.h>

// ---------------- problem constants (match reference config) ----------------
#define BB   2
#define TT   128
#define NN   128
#define DD   512
#define KK   8
#define FF   48          // NUM_DIRS * (3 quantiles + poly_order + 1)
#define HW   16          // head width
#define MROWS (BB*TT*NN)         // 32768
#define HELEMS ((size_t)MROWS*DD) // 16777216 floats of h

// softsort reg = max(1e-3, 0.5 * 128^-0.5)
#define SS_REG 0.04419417382415922f

typedef __attribute__((ext_vector_type(2))) float v2f;
typedef __attribute__((ext_vector_type(8))) float v8f;

// ---------------- workspace layout (float offsets) ----------------
#define UPAD_OFF 0                   // 512*16 = 8192   padded U [D][16]
#define V_OFF    8192                // B*N*K*T = 262144 (v in [B,N,K,T] layout)
#define FE_OFF   270336              // B*N*F = 12288   feats
#define BY_OFF   282624              // B*N*D = 131072  beta_eff * y
#define SQ_OFF   413696              // B*F = 96        per-feature energy sums
#define ACC_OFF  413792              // [0]=num_sum, [1]=den_sum
#define WS_FLOATS 413794

// =====================================================================
// kernel 0: zero the small accumulators (sq[96] + acc[2] are contiguous)
// =====================================================================
__global__ void k_zero(float* __restrict__ sq_and_acc) {
    int i = threadIdx.x;
    if (i < 98) sq_and_acc[i] = 0.0f;
}

// =====================================================================
// kernel 1: modified Gram-Schmidt QR of W_proj^T (512x8) -> Upad [512][16]
// =====================================================================
__global__ void k_qr(const float* __restrict__ Wp, float* __restrict__ Upad) {
    __shared__ float q[KK][DD];
    __shared__ float red[DD];
    int tid = threadIdx.x;   // 512 threads, one per d
    for (int k = 0; k < KK; ++k) q[k][tid] = Wp[k * DD + tid];
    __syncthreads();
    for (int k = 0; k < KK; ++k) {
        for (int j = 0; j < k; ++j) {
            red[tid] = q[j][tid] * q[k][tid]; __syncthreads();
            for (int s = DD / 2; s > 0; s >>= 1) {
                if (tid < s) red[tid] += red[tid + s];
                __syncthreads();
            }
            float dot = red[0]; __syncthreads();
            q[k][tid] -= dot * q[j][tid]; __syncthreads();
        }
        red[tid] = q[k][tid] * q[k][tid]; __syncthreads();
        for (int s = DD / 2; s > 0; s >>= 1) {
            if (tid < s) red[tid] += red[tid + s];
            __syncthreads();
        }
        float inv = rsqrtf(fmaxf(red[0], 1e-20f)); __syncthreads();
        q[k][tid] *= inv; __syncthreads();
    }
    for (int k = 0; k < 16; ++k)
        Upad[tid * 16 + k] = (k < KK) ? q[k][tid] : 0.0f;
}

// =====================================================================
// kernel 2: v = (x @ U) * mask via V_WMMA_F32_16X16X4_F32, fused with the
// per-row ||x|| reduction (den). One wave per 16-row tile; x read once.
//   A frag (16x4 f32): lane<16 -> (K0,K1) of row lane; lane>=16 -> (K2,K3)
//   B frag (4x16 f32): lane<16 -> rows (0,1) col lane; lane>=16 -> rows (2,3)
//   C/D (16x16 f32):  lane<16 vgpr r -> M=r, N=lane; lane>=16 -> M=8+r
// =====================================================================
__global__ void k_proj(const float* __restrict__ x,
                       const float* __restrict__ vm,
                       const float* __restrict__ Upad,
                       float* __restrict__ vout,
                       float* __restrict__ acc) {
    __shared__ float uL[DD * 16];            // 32 KB padded U
    int tid = threadIdx.x;
    for (int i = tid; i < DD * 16; i += 256) uL[i] = Upad[i];
    __syncthreads();

    int wave = tid >> 5, lane = tid & 31;
    int tile = blockIdx.x * 8 + wave;        // 2048 tiles total
    int m0   = tile * 16;
    int hi   = lane >> 4;                    // 0|1 : half-lane group
    int laneM = lane & 15;

    const float* xrow = x + (size_t)(m0 + laneM) * DD + hi * 2;
    v8f c = {};
    float ss = 0.0f;
    #pragma unroll 4
    for (int ks = 0; ks < DD / 4; ++ks) {
        v2f a = *(const v2f*)(xrow + ks * 4);
        int rb = ks * 4 + hi * 2;
        v2f bf;
        bf.x = uL[rb * 16 + laneM];
        bf.y = uL[rb * 16 + 16 + laneM];
        ss += a.x * a.x + a.y * a.y;
        c = __builtin_amdgcn_wmma_f32_16x16x4_f32(
                false, a, false, bf, (short)0, c, false, false);
    }

    // row norms for den: lane pairs (L, L+16) cover one full row
    ss += __shfl_xor(ss, 16);
    float rn = (lane < 16) ? sqrtf(ss) : 0.0f;
    for (int o = 16; o > 0; o >>= 1) rn += __shfl_xor(rn, o);
    if (lane == 0) atomicAdd(&acc[1], rn);

    // scatter masked v into [B,N,K,T] layout
    int b  = m0 >> 14;            // / (T*N)
    int t  = (m0 >> 7) & (NN - 1);
    int n0 = m0 & (NN - 1);
    int kcol = laneM;
    if (kcol < KK) {
        #pragma unroll
        for (int r = 0; r < 8; ++r) {
            int M = r + hi * 8;
            int grow = m0 + M;                 // flat (b,t,n) row
            float val = c[r] * vm[grow];
            int n = n0 + M;
            vout[(((b * NN + n) * KK + kcol) * TT) + t] = val;
        }
    }
}

// =====================================================================
// kernel 3: per-sequence soft-sort (sort + exact PAV isotonic) quantiles
// plus normalized-Legendre coefficients -> feats [B*N, 48]
// one block of 128 threads per sequence (2048 blocks)
// =====================================================================
__global__ void k_feats(const float* __restrict__ vws,
                        float* __restrict__ feats) {
    __shared__ float s[TT];
    __shared__ float red[TT];
    __shared__ float pv[TT];
    __shared__ float pc[TT];
    __shared__ float fit[TT];
    __shared__ float outv[6];

    int tid = threadIdx.x;
    int seq = blockIdx.x;
    float v = vws[(size_t)seq * TT + tid];

    auto reduce = [&](float val) -> float {
        red[tid] = val; __syncthreads();
        for (int st = TT / 2; st > 0; st >>= 1) {
            if (tid < st) red[tid] += red[tid + st];
            __syncthreads();
        }
        float r = red[0]; __syncthreads();
        return r;
    };

    // Legendre basis values at this t, then normalized projections
    float tl = -1.0f + 2.0f * (float)tid / (float)(TT - 1);
    float P1 = tl;
    float P2 = 0.5f * (3.0f * tl * tl - 1.0f);
    float n1 = sqrtf(fmaxf(reduce(P1 * P1), 1e-6f));
    float n2 = sqrtf(fmaxf(reduce(P2 * P2), 1e-6f));
    float c0 = reduce(v)       / sqrtf((float)TT);
    float c1 = reduce(v * P1)  / n1;
    float c2 = reduce(v * P2)  / n2;

    // bitonic sort ascending (128 elements in LDS)
    s[tid] = v; __syncthreads();
    for (int ksz = 2; ksz <= TT; ksz <<= 1) {
        for (int j = ksz >> 1; j > 0; j >>= 1) {
            int ixj = tid ^ j;
            if (ixj > tid) {
                float a = s[tid], b = s[ixj];
                bool up = ((tid & ksz) == 0);
                if ((a > b) == up) { s[tid] = b; s[ixj] = a; }
            }
            __syncthreads();
        }
    }
    // z_t = rho_t + sorted_asc_t, rho_t = (T - t)/reg  (descending ramp)
    s[tid] = (float)(TT - tid) / SS_REG + s[tid];
    __syncthreads();

    if (tid == 0) {
        // exact L2 isotonic regression, nonincreasing (PAV)
        int np = 0;
        for (int t2 = 0; t2 < TT; ++t2) {
            float val = s[t2], cnt = 1.0f;
            while (np > 0 && pv[np - 1] <= val) {
                val = (pv[np - 1] * pc[np - 1] + val * cnt) / (pc[np - 1] + cnt);
                cnt += pc[np - 1];
                --np;
            }
            pv[np] = val; pc[np] = cnt; ++np;
        }
        int idx = 0;
        for (int p = 0; p < np; ++p) {
            int cc = (int)pc[p];
            for (int q = 0; q < cc; ++q) fit[idx++] = pv[p];
        }
        // ss_t = fit_t - rho_t ; interpolated quantiles at 0.1/0.5/0.9
        const float lvl[3] = {0.1f, 0.5f, 0.9f};
        for (int q = 0; q < 3; ++q) {
            float pos = lvl[q] * (float)(TT - 1);
            int lo = (int)floorf(pos);
            int hiI = min(lo + 1, TT - 1);
            float w = pos - (float)lo;
            float sslo = fit[lo]  - (float)(TT - lo)  / SS_REG;
            float sshi = fit[hiI] - (float)(TT - hiI) / SS_REG;
            outv[q] = (1.0f - w) * sslo + w * sshi;
        }
        outv[3] = c0; outv[4] = c1; outv[5] = c2;
    }
    __syncthreads();
    if (tid < 6) {
        int k = seq & (KK - 1), bn = seq >> 3;
        feats[bn * FF + k * 6 + tid] = outv[tid];
    }
}

// =====================================================================
// kernel 4: per-(b,n) additive head: gelu MLP, y, beta*y, energy stats
// one block of 256 threads per (b,n) (256 blocks)
// =====================================================================
__global__ void k_head(const float* __restrict__ feats,
                       const float* __restrict__ W1,
                       const float* __restrict__ b1,
                       const float* __restrict__ W2,
                       const float* __restrict__ beta,
                       float* __restrict__ byOut,
                       float* __restrict__ sqG,
                       float* __restrict__ acc) {
    __shared__ float fL[FF];
    __shared__ float hm[FF * HW];
    __shared__ float sqL[FF];
    __shared__ float red[256];
    int tid = threadIdx.x;
    int bn = blockIdx.x;
    int b = bn >> 7;

    if (tid < FF) { fL[tid] = feats[bn * FF + tid]; sqL[tid] = 0.0f; }
    __syncthreads();
    for (int i = tid; i < FF * HW; i += 256) {
        int f = i >> 4;
        float pre = fL[f] * W1[i] + b1[i];
        hm[i] = 0.5f * pre * (1.0f + erff(pre * 0.70710678118654752f));
    }
    __syncthreads();

    float nacc = 0.0f;
    for (int dd = 0; dd < 2; ++dd) {
        int d = tid + dd * 256;
        float bd = 1.0f / (1.0f + expf(-beta[d]));     // sigmoid(beta)
        float ysum = 0.0f;
        for (int f = 0; f < FF; ++f) {
            float pf = 0.0f;
            #pragma unroll
            for (int w = 0; w < HW; ++w)
                pf += hm[f * HW + w] * W2[(size_t)(f * HW + w) * DD + d];
            ysum += pf;
            float bp = bd * pf;
            atomicAdd(&sqL[f], bp * bp);               // ds_add_f32
        }
        float by = bd * ysum;
        byOut[(size_t)bn * DD + d] = by;
        nacc += by * by;
    }
    // ||beta*y|| for this (b,n) -> num accumulator
    red[tid] = nacc; __syncthreads();
    for (int st = 128; st > 0; st >>= 1) {
        if (tid < st) red[tid] += red[tid + st];
        __syncthreads();
    }
    if (tid == 0) atomicAdd(&acc[0], sqrtf(red[0]));
    __syncthreads();
    if (tid < FF) atomicAdd(&sqG[b * FF + tid], sqL[tid]);
}

// =====================================================================
// kernel 5: fused broadcast output  h = x + vm * (beta*y)   (float4 pass)
// =====================================================================
__global__ void k_out(const float* __restrict__ x,
                      const float* __restrict__ vm,
                      const float* __restrict__ byWs,
                      float* __restrict__ hOut) {
    size_t i4 = (size_t)blockIdx.x * 256 + threadIdx.x;   // over HELEMS/4
    float4 x4 = ((const float4*)x)[i4];
    int d4 = (int)(i4 & (DD / 4 - 1));
    size_t rowm = i4 >> 7;                                // flat (b,t,n)
    int n = (int)(rowm & (NN - 1));
    int b = (int)(rowm >> 14);
    float m = vm[rowm];
    float4 by4 = ((const float4*)byWs)[(size_t)(b * NN + n) * (DD / 4) + d4];
    float4 h;
    h.x = x4.x + m * by4.x;
    h.y = x4.y + m * by4.y;
    h.z = x4.z + m * by4.z;
    h.w = x4.w + m * by4.w;
    ((float4*)hOut)[i4] = h;
}

// =====================================================================
// kernel 6: finalize scalars r and r_feat into the output tail
// =====================================================================
__global__ void k_fin(const float* __restrict__ acc,
                      const float* __restrict__ sqG,
                      float* __restrict__ out) {
    int tid = threadIdx.x;
    float den = fmaxf(acc[1] / (float)MROWS, 1e-9f);
    if (tid == 0) out[HELEMS] = (acc[0] / (float)(BB * NN)) / den;
    if (tid < FF) {
        float e = 0.5f * (sqrtf((float)TT * sqG[tid]) +
                          sqrtf((float)TT * sqG[FF + tid]));
        out[HELEMS + 1 + tid] = e / den;
    }
}

// =====================================================================
extern "C" void kernel_launch(void* const* d_in, const int* in_sizes, int n_in,
                              void* d_out, int out_size, void* d_ws, size_t ws_size,
                              hipStream_t stream) {
    const float* x    = (const float*)d_in[0];
    const float* vm   = (const float*)d_in[1];
    const float* Wp   = (const float*)d_in[2];
    const float* W1   = (const float*)d_in[3];
    const float* b1   = (const float*)d_in[4];
    const float* W2   = (const float*)d_in[5];
    const float* beta = (const float*)d_in[6];

    float* ws    = (float*)d_ws;
    float* Upad  = ws + UPAD_OFF;
    float* vws   = ws + V_OFF;
    float* feats = ws + FE_OFF;
    float* byWs  = ws + BY_OFF;
    float* sqG   = ws + SQ_OFF;
    float* acc   = ws + ACC_OFF;
    float* out   = (float*)d_out;

    k_zero <<<1, 128, 0, stream>>>(sqG);                        // zeros sq[96]+acc[2]
    k_qr   <<<1, DD, 0, stream>>>(Wp, Upad);
    k_proj <<<MROWS / 16 / 8, 256, 0, stream>>>(x, vm, Upad, vws, acc);
    k_feats<<<BB * NN * KK, TT, 0, stream>>>(vws, feats);
    k_head <<<BB * NN, 256, 0, stream>>>(feats, W1, b1, W2, beta, byWs, sqG, acc);
    k_out  <<<(int)(HELEMS / 4 / 256), 256, 0, stream>>>(x, vm, byWs, out);
    k_fin  <<<1, 64, 0, stream>>>(acc, sqG, out);
}